// MambaBackbone_80762565034367
// MI455X (gfx1250) — compile-verified
//
#include <hip/hip_runtime.h>

typedef float v2f __attribute__((ext_vector_type(2)));
typedef float v4f __attribute__((ext_vector_type(4)));
typedef float v8f __attribute__((ext_vector_type(8)));

#define N_DEPTH 12
#define N_B     512
#define D_MODEL 1024
#define D_INNER 2048
#define D_STATE 16
#define D_CONV  4
#define DT_RANK 64
#define D_IN    8

#define BM       16     // batch rows per workgroup == WMMA M-tile (fully utilized)
#define NTHREADS 512
#define NWAVES   16

#define HS   1028       // padded LDS stride for h   (1024+4 -> conflict-free b64 reads)
#define XS   2052       // padded LDS stride for x/y (2048+4)
#define XDBS 100        // padded LDS stride for xdb (96+4)

// One 16x16 fp32 output tile via V_WMMA_F32_16X16X4_F32, K-loop over LDS A and global B.
__device__ __forceinline__ v8f wmma_f32_tile(const float* __restrict__ Bw, int ldb,
                                             int n0, const float* sA, int lda,
                                             int K, int lane) {
  const int mr = lane & 15;            // A row (lanes 0-15: K={k0,k0+1}; 16-31: K={k0+2,k0+3})
  const int kb = (lane >> 4) << 1;
  const int nn = n0 + (lane & 15);
  const float* ap = sA + mr * lda + kb;
  const float* bp = Bw + (size_t)nn * ldb + kb;
  v8f acc = {};
  #pragma unroll 4
  for (int k0 = 0; k0 < K; k0 += 4) {
    v2f a = *(const v2f*)(ap + k0);    // ds_load_b64
    v2f b = *(const v2f*)(bp + k0);    // global_load_b64
    acc = __builtin_amdgcn_wmma_f32_16x16x4_f32(false, a, false, b, (short)0, acc,
                                                false, false);
  }
  return acc;
}

__global__ __launch_bounds__(NTHREADS, 1)
void mamba_fused(const float* __restrict__ x_t,
                 const float* __restrict__ conv_states,
                 const float* __restrict__ ssm_states,
                 const float* __restrict__ W_in,
                 const float* __restrict__ b_in,
                 const float* __restrict__ in_proj_w,
                 const float* __restrict__ conv_w,
                 const float* __restrict__ conv_b,
                 const float* __restrict__ x_proj_w,
                 const float* __restrict__ dt_proj_w,
                 const float* __restrict__ dt_proj_b,
                 const float* __restrict__ A_log,
                 const float* __restrict__ D_param,
                 const float* __restrict__ out_proj_w,
                 const float* __restrict__ ln_w,
                 const float* __restrict__ ln_b,
                 float* __restrict__ out) {
  __shared__ float lds[BM * HS + BM * XS + BM * XDBS];   // ~199 KB (<=320 KB/WG per ISA)
  float* sh_h   = lds;
  float* sh_x   = lds + BM * HS;
  float* sh_xdb = lds + BM * HS + BM * XS;

  const int tid  = threadIdx.x;
  const int lane = tid & 31;
  const int wave = tid >> 5;
  const int b0   = blockIdx.x * BM;

  // ---- Phase 0: h = x_t @ W_in^T + b_in  (K=8, trivial) ----
  for (int e = tid; e < BM * D_MODEL; e += NTHREADS) {
    int r = e >> 10, c = e & (D_MODEL - 1);
    float s = b_in[c];
    const float* xr = x_t + (size_t)(b0 + r) * D_IN;
    const float* wr = W_in + (size_t)c * D_IN;
    #pragma unroll
    for (int k = 0; k < D_IN; ++k) s += xr[k] * wr[k];
    sh_h[r * HS + c] = s;
  }
  __syncthreads();

  for (int l = 0; l < N_DEPTH; ++l) {
    const float* ipw = in_proj_w  + (size_t)l * (2 * D_INNER) * D_MODEL;
    const float* cw  = conv_w     + (size_t)l * D_INNER * D_CONV;
    const float* cb  = conv_b     + (size_t)l * D_INNER;
    const float* xpw = x_proj_w   + (size_t)l * (DT_RANK + 2 * D_STATE) * D_INNER;
    const float* dtw = dt_proj_w  + (size_t)l * D_INNER * DT_RANK;
    const float* dtb = dt_proj_b  + (size_t)l * D_INNER;
    const float* al  = A_log      + (size_t)l * D_INNER * D_STATE;
    const float* dp  = D_param    + (size_t)l * D_INNER;
    const float* opw = out_proj_w + (size_t)l * D_MODEL * D_INNER;
    const float* cst = conv_states + (size_t)(l * N_B + b0) * D_INNER * D_CONV;
    const float* sst = ssm_states  + (size_t)(l * N_B + b0) * D_INNER * D_STATE;

    // ---- Phase A: raw x = h @ ipw_x^T  (M16 N2048 K1024, WMMA) ----
    for (int t = wave; t < D_INNER / 16; t += NWAVES) {
      v8f acc = wmma_f32_tile(ipw, D_MODEL, t * 16, sh_h, HS, D_MODEL, lane);
      int n = t * 16 + (lane & 15);
      int mb = (lane >> 4) << 3;       // lanes 0-15 hold rows 0..7; 16-31 rows 8..15
      #pragma unroll
      for (int i = 0; i < 8; ++i) sh_x[(mb + i) * XS + n] = acc[i];
    }
    __syncthreads();

    // ---- Phase A2: causal-conv mix + SiLU (states read once -> non-temporal) ----
    for (int e = tid; e < BM * D_INNER; e += NTHREADS) {
      int r = e >> 11, d = e & (D_INNER - 1);
      v4f c4 = __builtin_nontemporal_load(
          (const v4f*)(cst + ((size_t)r * D_INNER + d) * D_CONV));
      const float* w4 = cw + d * D_CONV;
      float xv = sh_x[r * XS + d];
      float s = c4[1] * w4[0] + c4[2] * w4[1] + c4[3] * w4[2] + xv * w4[3] + cb[d];
      sh_x[r * XS + d] = s / (1.f + __expf(-s));
    }
    __syncthreads();

    // ---- Phase B: xdb = x @ xpw^T  (N=96 -> 6 WMMA tiles) ----
    if (wave < 6) {
      v8f acc = wmma_f32_tile(xpw, D_INNER, wave * 16, sh_x, XS, D_INNER, lane);
      int n = wave * 16 + (lane & 15);
      int mb = (lane >> 4) << 3;
      #pragma unroll
      for (int i = 0; i < 8; ++i) sh_xdb[(mb + i) * XDBS + n] = acc[i];
    }
    __syncthreads();

    // ---- Phase C: dt (fused K=64 dot) + softplus + selective scan; y_raw -> sh_x ----
    {
      const int r = wave;               // one wave per batch row (16 waves, 16 rows)
      float Bm[D_STATE], Cm[D_STATE];
      #pragma unroll
      for (int n = 0; n < D_STATE; ++n) {
        Bm[n] = sh_xdb[r * XDBS + DT_RANK + n];
        Cm[n] = sh_xdb[r * XDBS + DT_RANK + D_STATE + n];
      }
      for (int j = 0; j < D_INNER / 32; ++j) {
        int d = j * 32 + lane;          // coalesced across lanes
        float s = dtb[d];
        const float* dw = dtw + (size_t)d * DT_RANK;
        #pragma unroll 8
        for (int k = 0; k < DT_RANK; ++k) s += sh_xdb[r * XDBS + k] * dw[k];
        float dt = (s > 20.f) ? s : __logf(1.f + __expf(s));
        float xv = sh_x[r * XS + d];
        float dtx = dt * xv;
        const float* ssp = sst + ((size_t)r * D_INNER + d) * D_STATE;
        const float* alp = al + (size_t)d * D_STATE;
        float y = 0.f;
        #pragma unroll
        for (int n4 = 0; n4 < 4; ++n4) {
          v4f sv = __builtin_nontemporal_load((const v4f*)(ssp + 4 * n4));
          v4f av = *(const v4f*)(alp + 4 * n4);
          #pragma unroll
          for (int q = 0; q < 4; ++q) {
            int n = 4 * n4 + q;
            float dA = __expf(-__expf(av[q]) * dt);
            float sn = sv[q] * dA + dtx * Bm[n];
            y += sn * Cm[n];
          }
        }
        y += dp[d] * xv;
        sh_x[r * XS + d] = y;
      }
    }
    __syncthreads();

    // ---- Phase D: z = h @ ipw_z^T (deferred half of in_proj), fused y *= silu(z) ----
    const float* ipwz = ipw + (size_t)D_INNER * D_MODEL;
    for (int t = wave; t < D_INNER / 16; t += NWAVES) {
      v8f acc = wmma_f32_tile(ipwz, D_MODEL, t * 16, sh_h, HS, D_MODEL, lane);
      int n = t * 16 + (lane & 15);
      int mb = (lane >> 4) << 3;
      #pragma unroll
      for (int i = 0; i < 8; ++i) {
        float zv = acc[i];
        int idx = (mb + i) * XS + n;
        sh_x[idx] = sh_x[idx] * (zv / (1.f + __expf(-zv)));
      }
    }
    __syncthreads();

    // ---- Phase E: h = y @ opw^T (M16 N1024 K2048, WMMA) ----
    for (int t = wave; t < D_MODEL / 16; t += NWAVES) {
      v8f acc = wmma_f32_tile(opw, D_INNER, t * 16, sh_x, XS, D_INNER, lane);
      int n = t * 16 + (lane & 15);
      int mb = (lane >> 4) << 3;
      #pragma unroll
      for (int i = 0; i < 8; ++i) sh_h[(mb + i) * HS + n] = acc[i];
    }
    __syncthreads();
  }

  // ---- Final LayerNorm, one wave per row ----
  {
    const int r = wave;
    float s = 0.f, s2 = 0.f;
    for (int c = lane; c < D_MODEL; c += 32) {
      float v = sh_h[r * HS + c];
      s += v; s2 += v * v;
    }
    #pragma unroll
    for (int off = 16; off > 0; off >>= 1) {
      s  += __shfl_xor(s, off, 32);
      s2 += __shfl_xor(s2, off, 32);
    }
    float mu   = s * (1.f / D_MODEL);
    float var  = s2 * (1.f / D_MODEL) - mu * mu;
    float rinv = rsqrtf(var + 1e-5f);
    for (int c = lane; c < D_MODEL; c += 32) {
      float v = sh_h[r * HS + c];
      out[(size_t)(b0 + r) * D_MODEL + c] = (v - mu) * rinv * ln_w[c] + ln_b[c];
    }
  }
}

extern "C" void kernel_launch(void* const* d_in, const int* in_sizes, int n_in,
                              void* d_out, int out_size, void* d_ws, size_t ws_size,
                              hipStream_t stream) {
  const float* x_t         = (const float*)d_in[0];
  const float* conv_states = (const float*)d_in[1];
  const float* ssm_states  = (const float*)d_in[2];
  const float* W_in        = (const float*)d_in[3];
  const float* b_in        = (const float*)d_in[4];
  const float* in_proj_w   = (const float*)d_in[5];
  const float* conv_w      = (const float*)d_in[6];
  const float* conv_b      = (const float*)d_in[7];
  const float* x_proj_w    = (const float*)d_in[8];
  const float* dt_proj_w   = (const float*)d_in[9];
  const float* dt_proj_b   = (const float*)d_in[10];
  const float* A_log       = (const float*)d_in[11];
  const float* D_param     = (const float*)d_in[12];
  const float* out_proj_w  = (const float*)d_in[13];
  const float* ln_w        = (const float*)d_in[14];
  const float* ln_b        = (const float*)d_in[15];
  float* out = (float*)d_out;

  mamba_fused<<<N_B / BM, NTHREADS, 0, stream>>>(
      x_t, conv_states, ssm_states, W_in, b_in, in_proj_w, conv_w, conv_b,
      x_proj_w, dt_proj_w, dt_proj_b, A_log, D_param, out_proj_w, ln_w, ln_b, out);
}